// GCNModel_7215545057639
// MI455X (gfx1250) — compile-verified
//
#include <hip/hip_runtime.h>
#include <math.h>

typedef __attribute__((ext_vector_type(2))) float v2f;
typedef __attribute__((ext_vector_type(8))) float v8f;

#define NN 100000

// ---------------- utility: zero a float region ----------------
__global__ void zero_f32(float* __restrict__ p, int n4) {
    int i = blockIdx.x * blockDim.x + threadIdx.x;
    int stride = gridDim.x * blockDim.x;
    float4 z = make_float4(0.f, 0.f, 0.f, 0.f);
    for (; i < n4; i += stride) ((float4*)p)[i] = z;
}

// ---------------- degree counting ----------------
__global__ void deg_kernel(const int* __restrict__ src, const int* __restrict__ dst,
                           unsigned* __restrict__ out_cnt, unsigned* __restrict__ in_cnt, int E) {
    int e = blockIdx.x * blockDim.x + threadIdx.x;
    if (e < E) {
        atomicAdd(&out_cnt[src[e]], 1u);
        atomicAdd(&in_cnt[dst[e]], 1u);
    }
}

__global__ void norm_kernel(const unsigned* __restrict__ out_cnt, const unsigned* __restrict__ in_cnt,
                            float* __restrict__ out_norm, float* __restrict__ in_norm, int n) {
    int i = blockIdx.x * blockDim.x + threadIdx.x;
    if (i < n) {
        // +1 for self loop; degree >= 1 so clip is implicit
        out_norm[i] = rsqrtf((float)(out_cnt[i] + 1u));
        in_norm[i]  = rsqrtf((float)(in_cnt[i] + 1u));
    }
}

// ---------------- h = weight[:,None] @ W_in + b_in ----------------
__global__ void init_h_kernel(const float* __restrict__ weight, const float* __restrict__ W_in,
                              const float* __restrict__ b_in, float* __restrict__ h, int n) {
    int idx = blockIdx.x * blockDim.x + threadIdx.x;   // over n*16 float4 chunks
    if (idx >= n * 16) return;
    int node = idx >> 4;
    int q = idx & 15;
    float w = weight[node];
    float4 wi = ((const float4*)W_in)[q];
    float4 bi = ((const float4*)b_in)[q];
    float4 r;
    r.x = w * wi.x + bi.x;
    r.y = w * wi.y + bi.y;
    r.z = w * wi.z + bi.z;
    r.w = w * wi.w + bi.w;
    ((float4*)(h + (size_t)node * 64))[q] = r;
}

// ---------------- fused activation*out_norm + [N,64]x[64,64] GEMM via f32 WMMA ----------------
// block = 256 threads (8 waves); block tile = 32 rows x 64 cols
// wave w: m_sub = w>>2 (0..1), n_sub = w&3 (0..3) -> one 16x16 output tile per wave
__global__ __launch_bounds__(256) void gemm_layer_kernel(
    const float* __restrict__ h, const float* __restrict__ W,
    const float* __restrict__ out_norm, float* __restrict__ g, int act)
{
    __shared__ float Wl[64 * 64];   // 16 KB
    __shared__ float Tl[32 * 64];   // 8 KB  (activation+norm applied input tile)

    const int tid = threadIdx.x;
    const int row0 = blockIdx.x * 32;

    // stage W (4096 floats = 1024 float4)
    for (int i = tid; i < 1024; i += 256)
        ((float4*)Wl)[i] = ((const float4*)W)[i];

    // stage input tile with activation and out_norm scaling (512 float4)
    for (int i = tid; i < 512; i += 256) {
        int r  = i >> 4;          // 0..31
        int c4 = i & 15;          // float4 index within row
        int row = row0 + r;
        float4 v = ((const float4*)(h + (size_t)row * 64))[c4];
        if (act) {
            v.x = v.x > 0.f ? v.x : 0.01f * v.x;
            v.y = v.y > 0.f ? v.y : 0.01f * v.y;
            v.z = v.z > 0.f ? v.z : 0.01f * v.z;
            v.w = v.w > 0.f ? v.w : 0.01f * v.w;
        }
        float s = out_norm[row];
        v.x *= s; v.y *= s; v.z *= s; v.w *= s;
        ((float4*)(Tl + r * 64))[c4] = v;
    }
    __syncthreads();

    const int wave = tid >> 5;
    const int lane = tid & 31;
    const int msub = wave >> 2;
    const int nsub = wave & 3;
    const int l16  = lane & 15;
    const int arow = msub * 16 + l16;              // A row in Tl
    const int koff = (lane < 16) ? 0 : 2;          // A/B lane K offset per ISA layout
    const int col  = nsub * 16 + l16;              // B/D column

    v8f c = {};
    #pragma unroll
    for (int kk = 0; kk < 16; ++kk) {
        int k = kk * 4 + koff;
        v2f a, b;
        // A 16x4: v0 = K=k, v1 = K=k+1 for this lane's row
        a.x = Tl[arow * 64 + k];
        a.y = Tl[arow * 64 + k + 1];
        // B 4x16: v0 = row K=k, v1 = row K=k+1, at column `col`
        b.x = Wl[k * 64 + col];
        b.y = Wl[(k + 1) * 64 + col];
        c = __builtin_amdgcn_wmma_f32_16x16x4_f32(false, a, false, b,
                                                  (short)0, c, false, false);
    }

    // D 16x16 layout: VGPR r -> row M=r (lanes 0-15) / M=8+r (lanes 16-31), N=lane%16
    const int rbase = row0 + msub * 16 + ((lane < 16) ? 0 : 8);
    #pragma unroll
    for (int r = 0; r < 8; ++r)
        g[(size_t)(rbase + r) * 64 + col] = c[r];
}

// ---------------- SpMM scatter: one wave per edge, 2 f32 atomics/lane ----------------
__global__ __launch_bounds__(256) void spmm_kernel(
    const float* __restrict__ g, const int* __restrict__ src, const int* __restrict__ dst,
    float* __restrict__ agg, int E)
{
    int e = blockIdx.x * 8 + (threadIdx.x >> 5);
    if (e >= E) return;
    int lane = threadIdx.x & 31;
    int s = src[e];
    int d = dst[e];
    float2 v = ((const float2*)(g + (size_t)s * 64))[lane];
    float* a = agg + (size_t)d * 64 + lane * 2;
    atomicAdd(a,     v.x);
    atomicAdd(a + 1, v.y);
}

// ---------------- epilogue: h = (agg + g_selfloop) * in_norm + bias ----------------
__global__ void post_kernel(const float* __restrict__ agg, const float* __restrict__ g,
                            const float* __restrict__ in_norm, const float* __restrict__ bc,
                            float* __restrict__ h, int n) {
    int idx = blockIdx.x * blockDim.x + threadIdx.x;   // over n*16 float4 chunks
    if (idx >= n * 16) return;
    int node = idx >> 4;
    int q = idx & 15;
    float s = in_norm[node];
    float4 a = ((const float4*)(agg + (size_t)node * 64))[q];
    float4 gv = ((const float4*)(g + (size_t)node * 64))[q];
    float4 b = ((const float4*)bc)[q];
    float4 r;
    r.x = (a.x + gv.x) * s + b.x;
    r.y = (a.y + gv.y) * s + b.y;
    r.z = (a.z + gv.z) * s + b.z;
    r.w = (a.w + gv.w) * s + b.w;
    ((float4*)(h + (size_t)node * 64))[q] = r;
}

// ---------------- prediction head: logits = h @ W_pred + b_pred ----------------
__global__ __launch_bounds__(256) void pred_kernel(
    const float* __restrict__ h, const float* __restrict__ W_pred,
    const float* __restrict__ b_pred, float* __restrict__ out, int n)
{
    int node = blockIdx.x * 8 + (threadIdx.x >> 5);
    if (node >= n) return;
    int lane = threadIdx.x & 31;
    float2 w = ((const float2*)W_pred)[lane];
    float2 v = ((const float2*)(h + (size_t)node * 64))[lane];
    float s = v.x * w.x + v.y * w.y;
    for (int off = 16; off > 0; off >>= 1)
        s += __shfl_down(s, off, 32);
    if (lane == 0) out[node] = s + b_pred[0];
}

extern "C" void kernel_launch(void* const* d_in, const int* in_sizes, int n_in,
                              void* d_out, int out_size, void* d_ws, size_t ws_size,
                              hipStream_t stream) {
    const float* weight = (const float*)d_in[0];   // [N]
    const int*   src    = (const int*)d_in[1];     // [E]
    const int*   dst    = (const int*)d_in[2];     // [E]
    const float* W_in   = (const float*)d_in[3];   // [1,64]
    const float* b_in   = (const float*)d_in[4];   // [64]
    const float* Wc     = (const float*)d_in[5];   // [3,64,64]
    const float* bc     = (const float*)d_in[6];   // [3,64]
    const float* W_pred = (const float*)d_in[7];   // [64,1]
    const float* b_pred = (const float*)d_in[8];   // [1]
    float* out = (float*)d_out;

    const int N = in_sizes[0];
    const int E = in_sizes[1];

    // workspace layout
    unsigned* out_cnt = (unsigned*)d_ws;                 // N
    unsigned* in_cnt  = out_cnt + N;                     // N
    float* out_norm   = (float*)(in_cnt + N);            // N
    float* in_norm    = out_norm + N;                    // N
    float* h          = in_norm + N;                     // N*64
    float* g          = h + (size_t)N * 64;              // N*64
    float* agg        = g + (size_t)N * 64;              // N*64

    // 1) zero degree counters (2N u32 == zero-bit floats)
    zero_f32<<<512, 256, 0, stream>>>((float*)out_cnt, (2 * N) / 4);

    // 2) degrees
    deg_kernel<<<(E + 255) / 256, 256, 0, stream>>>(src, dst, out_cnt, in_cnt, E);

    // 3) norms
    norm_kernel<<<(N + 255) / 256, 256, 0, stream>>>(out_cnt, in_cnt, out_norm, in_norm, N);

    // 4) h = weight outer W_in + b_in
    init_h_kernel<<<(N * 16 + 255) / 256, 256, 0, stream>>>(weight, W_in, b_in, h, N);

    // 5) three GraphConv layers
    for (int l = 0; l < 3; ++l) {
        gemm_layer_kernel<<<N / 32, 256, 0, stream>>>(h, Wc + (size_t)l * 64 * 64,
                                                      out_norm, g, l > 0 ? 1 : 0);
        zero_f32<<<2048, 256, 0, stream>>>(agg, (N * 64) / 4);
        spmm_kernel<<<(E + 7) / 8, 256, 0, stream>>>(g, src, dst, agg, E);
        post_kernel<<<(N * 16 + 255) / 256, 256, 0, stream>>>(agg, g, in_norm,
                                                              bc + (size_t)l * 64, h, N);
    }

    // 6) prediction head
    pred_kernel<<<(N + 7) / 8, 256, 0, stream>>>(h, W_pred, b_pred, out, N);
}